// QuantumCritic_50921132261798
// MI455X (gfx1250) — compile-verified
//
#include <hip/hip_runtime.h>

typedef __attribute__((ext_vector_type(2))) float v2f;
typedef __attribute__((ext_vector_type(8))) float v8f;

#define NQ    8
#define NANG  24   // 3*NQ data-dependent angles
#define NLAY  2

// ---------------------------------------------------------------------------
// Kernel 1: angles = atan(x @ W_adapter^T + b_adapter) via V_WMMA_F32_16X16X4_F32
//   One wave computes a 16(M) x 24(N) tile, K=64 as 16 steps of K=4.
//   A frag (16x4 f32, 2 VGPR): lane m=lane%16, k = 2*(lane/16) + i
//   B frag (4x16 f32, 2 VGPR): lane n=lane%16, k = 2*(lane/16) + i
//   D (16x16 f32, 8 VGPR):     vgpr r -> m = r + 8*(lane/16), n = lane%16
//
//   N-tile 1 covers columns 16..31 but only 16..23 are real. Since WMMA
//   column n of D depends only on column n of B (the lanes with lane%16==n),
//   out-of-range lanes just load a clamped (valid) address and produce
//   garbage columns that are never stored -> inner loop is branch-free,
//   EXEC stays all-1s throughout (WMMA requirement), all loads are b64.
// ---------------------------------------------------------------------------
__global__ __launch_bounds__(256)
void adapter_wmma_kernel(const float* __restrict__ x,     // [B,64]
                         const float* __restrict__ W,     // [24,64]
                         const float* __restrict__ bias,  // [24]
                         float* __restrict__ angles)      // [B,24]
{
    const int lane  = threadIdx.x & 31;
    const int wave  = threadIdx.x >> 5;
    const int mtile = blockIdx.x * 8 + wave;
    const int m0    = mtile * 16;

    const int mrow  = lane & 15;          // A row / D col
    const int khalf = (lane >> 4) * 2;    // 0 or 2
    const int n0    = mrow;               // N-tile 0 column (always < 24)
    const int n1    = mrow + 16;          // N-tile 1 column (valid if < 24)
    const int n1c   = (n1 < NANG) ? n1 : (NANG - 1);  // clamped, always valid

    const float* xrow = x + (m0 + mrow) * 64 + khalf;
    const float* w0p  = W + n0  * 64 + khalf;
    const float* w1p  = W + n1c * 64 + khalf;

    v8f acc0 = {};
    v8f acc1 = {};

#pragma unroll
    for (int kk = 0; kk < 16; ++kk) {
        const int kb = kk * 4;
        v2f a, b0, b1;
        a.x  = xrow[kb + 0];
        a.y  = xrow[kb + 1];
        b0.x = w0p[kb + 0];
        b0.y = w0p[kb + 1];
        b1.x = w1p[kb + 0];
        b1.y = w1p[kb + 1];
        acc0 = __builtin_amdgcn_wmma_f32_16x16x4_f32(
                   false, a, false, b0, (short)0, acc0, false, false);
        acc1 = __builtin_amdgcn_wmma_f32_16x16x4_f32(
                   false, a, false, b1, (short)0, acc1, false, false);
    }

    const int mh = (lane >> 4) * 8;
    const float bias0 = bias[n0];
    const float bias1 = bias[n1c];
#pragma unroll
    for (int r = 0; r < 8; ++r) {
        const int m = m0 + r + mh;
        angles[m * NANG + n0] = atanf(acc0[r] + bias0);
        if (n1 < NANG)
            angles[m * NANG + n1] = atanf(acc1[r] + bias1);
    }
}

// ---------------------------------------------------------------------------
// Kernel 2: 8-qubit statevector simulation, one batch element per wave32.
//   Amplitude index a = (k<<5)|lane; k = register index 0..7, lane = 0..31.
//   Qubit q acts on bit (7-q): q=0..2 -> register bits 2..0 (intra-lane),
//   q=3..7 -> lane bits 4..0 (shfl_xor masks 16,8,4,2,1).
// ---------------------------------------------------------------------------
__device__ __forceinline__ void rx_gate(int q, float c, float s,
                                        float (&xr)[8], float (&xi)[8])
{
    if (q < 3) {
        const int m = 2 - q;
#pragma unroll
        for (int kk = 0; kk < 4; ++kk) {
            const int k0 = ((kk >> m) << (m + 1)) | (kk & ((1 << m) - 1));
            const int k1 = k0 | (1 << m);
            const float x0 = xr[k0], y0 = xi[k0];
            const float x1 = xr[k1], y1 = xi[k1];
            xr[k0] = c * x0 + s * y1;  xi[k0] = c * y0 - s * x1;
            xr[k1] = c * x1 + s * y0;  xi[k1] = c * y1 - s * x0;
        }
    } else {
        const int mask = 1 << (7 - q);
#pragma unroll
        for (int k = 0; k < 8; ++k) {
            const float pr = __shfl_xor(xr[k], mask, 32);
            const float pi = __shfl_xor(xi[k], mask, 32);
            const float x = xr[k], y = xi[k];
            xr[k] = c * x + s * pi;   // new = c*mine - i*s*partner (RX symmetric)
            xi[k] = c * y - s * pr;
        }
    }
}

__device__ __forceinline__ void ry_gate(int q, float c, float s,
                                        float (&xr)[8], float (&xi)[8], int lane)
{
    if (q < 3) {
        const int m = 2 - q;
#pragma unroll
        for (int kk = 0; kk < 4; ++kk) {
            const int k0 = ((kk >> m) << (m + 1)) | (kk & ((1 << m) - 1));
            const int k1 = k0 | (1 << m);
            const float x0 = xr[k0], y0 = xi[k0];
            const float x1 = xr[k1], y1 = xi[k1];
            xr[k0] = c * x0 - s * x1;  xi[k0] = c * y0 - s * y1;
            xr[k1] = s * x0 + c * x1;  xi[k1] = s * y0 + c * y1;
        }
    } else {
        const int mask = 1 << (7 - q);
        const float sg = (lane & mask) ? s : -s;
#pragma unroll
        for (int k = 0; k < 8; ++k) {
            const float pr = __shfl_xor(xr[k], mask, 32);
            const float pi = __shfl_xor(xi[k], mask, 32);
            xr[k] = c * xr[k] + sg * pr;
            xi[k] = c * xi[k] + sg * pi;
        }
    }
}

__device__ __forceinline__ void rz_gate(int q, float c, float s,
                                        float (&xr)[8], float (&xi)[8], int lane)
{
#pragma unroll
    for (int k = 0; k < 8; ++k) {
        const int a   = (k << 5) | lane;
        const int bit = (a >> (7 - q)) & 1;
        const float d = bit ? s : -s;            // phase e^{+-i t/2}
        const float x = xr[k], y = xi[k];
        xr[k] = x * c - y * d;
        xi[k] = y * c + x * d;
    }
}

__device__ __forceinline__ void cz_gate(int q, float (&xr)[8], float (&xi)[8], int lane)
{
#pragma unroll
    for (int k = 0; k < 8; ++k) {
        const int a = (k << 5) | lane;
        const float sg = (((a >> (7 - q)) & (a >> (6 - q)) & 1) != 0) ? -1.0f : 1.0f;
        xr[k] *= sg;
        xi[k] *= sg;
    }
}

__global__ __launch_bounds__(256)
void circuit_kernel(const float* __restrict__ angles, // [B,24]
                    const float* __restrict__ qw,     // [2,16] flat (32 vals)
                    const float* __restrict__ Wfc,    // [8]
                    const float* __restrict__ bfc,    // [1]
                    float* __restrict__ out)          // [B]
{
    const int lane = threadIdx.x & 31;
    const int wave = threadIdx.x >> 5;
    const int b    = blockIdx.x * 8 + wave;

    // Per-wave gate tables: lane g (g<24) holds cos/sin(angle_g/2);
    // lane l holds cos/sin(q_weights_flat[l]/2).
    float ac = 1.0f, asn = 0.0f;
    if (lane < NANG) {
        float s, c;
        sincosf(0.5f * angles[b * NANG + lane], &s, &c);
        ac = c; asn = s;
    }
    float qc, qs;
    {
        float s, c;
        sincosf(0.5f * qw[lane], &s, &c);
        qc = c; qs = s;
    }

    // Statevector |0...0>
    float xr[8], xi[8];
#pragma unroll
    for (int k = 0; k < 8; ++k) { xr[k] = 0.0f; xi[k] = 0.0f; }
    if (lane == 0) xr[0] = 1.0f;

#pragma unroll
    for (int l = 0; l < NLAY; ++l) {
#pragma unroll
        for (int q = 0; q < NQ; ++q) {
            float c, s;
            c = __shfl(ac, 3 * q,     32); s = __shfl(asn, 3 * q,     32);
            rx_gate(q, c, s, xr, xi);
            c = __shfl(ac, 3 * q + 1, 32); s = __shfl(asn, 3 * q + 1, 32);
            ry_gate(q, c, s, xr, xi, lane);
            c = __shfl(ac, 3 * q + 2, 32); s = __shfl(asn, 3 * q + 2, 32);
            rz_gate(q, c, s, xr, xi, lane);
        }
#pragma unroll
        for (int q = 0; q < NQ; ++q) {
            float c, s;
            c = __shfl(qc, l * 16 + q,      32); s = __shfl(qs, l * 16 + q,      32);
            ry_gate(q, c, s, xr, xi, lane);
            c = __shfl(qc, l * 16 + q + NQ, 32); s = __shfl(qs, l * 16 + q + NQ, 32);
            rz_gate(q, c, s, xr, xi, lane);
        }
#pragma unroll
        for (int q = 0; q < NQ - 1; ++q)
            cz_gate(q, xr, xi, lane);
    }

    // Readout: out = sum_a |amp_a|^2 * (sum_q Wfc[q]*Zsign_q(a)) + b_fc
    float w8[8];
#pragma unroll
    for (int q = 0; q < NQ; ++q) w8[q] = Wfc[q];

    float acc = 0.0f;
#pragma unroll
    for (int k = 0; k < 8; ++k) {
        const int a = (k << 5) | lane;
        const float p = xr[k] * xr[k] + xi[k] * xi[k];
        float w = 0.0f;
#pragma unroll
        for (int q = 0; q < NQ; ++q)
            w += (((a >> (7 - q)) & 1) != 0) ? -w8[q] : w8[q];
        acc += p * w;
    }
#pragma unroll
    for (int off = 16; off >= 1; off >>= 1)
        acc += __shfl_xor(acc, off, 32);

    if (lane == 0) out[b] = acc + bfc[0];
}

// ---------------------------------------------------------------------------
extern "C" void kernel_launch(void* const* d_in, const int* in_sizes, int n_in,
                              void* d_out, int out_size, void* d_ws, size_t ws_size,
                              hipStream_t stream)
{
    const float* x    = (const float*)d_in[0];   // [B,64]
    const float* Wad  = (const float*)d_in[1];   // [24,64]
    const float* bad  = (const float*)d_in[2];   // [24]
    const float* qw   = (const float*)d_in[3];   // [2,16]
    const float* Wfc  = (const float*)d_in[4];   // [1,8]
    const float* bfc  = (const float*)d_in[5];   // [1]
    float* out        = (float*)d_out;           // [B,1]

    const int B = in_sizes[0] / 64;              // 8192
    float* angles = (float*)d_ws;                // B*24 floats = 768 KB

    dim3 blk(256);
    // B/16 M-tiles, 8 waves per block
    adapter_wmma_kernel<<<dim3(B / 16 / 8), blk, 0, stream>>>(x, Wad, bad, angles);
    // one wave per batch element, 8 per block
    circuit_kernel<<<dim3(B / 8), blk, 0, stream>>>(angles, qw, Wfc, bfc, out);
}